// MTST_layer_22239340658731
// MI455X (gfx1250) — compile-verified
//
#include <hip/hip_runtime.h>
#include <math.h>

typedef __attribute__((ext_vector_type(2))) float v2f;
typedef __attribute__((ext_vector_type(8))) float v8f;

// ---------- wave helpers (wave32) ----------
__device__ __forceinline__ float wred_sum32(float v) {
#pragma unroll
  for (int off = 16; off >= 1; off >>= 1) v += __shfl_xor(v, off, 32);
  return v;
}
// reduce across the 16 "N" lanes (lanes 0-15 and 16-31 independently; xor masks < 16)
__device__ __forceinline__ float nred_sum(float v) {
#pragma unroll
  for (int off = 8; off >= 1; off >>= 1) v += __shfl_xor(v, off, 32);
  return v;
}
__device__ __forceinline__ float nred_max(float v) {
#pragma unroll
  for (int off = 8; off >= 1; off >>= 1) v = fmaxf(v, __shfl_xor(v, off, 32));
  return v;
}

__device__ __forceinline__ v8f wmma_f32(v2f a, v2f b, v8f c) {
  return __builtin_amdgcn_wmma_f32_16x16x4_f32(false, a, false, b, (short)0, c,
                                               false, false);
}

// ---------- patch extraction: out[b,i,j] = y_pad[b, i*stride + j] ----------
__global__ void mtst_patch(const float* __restrict__ y, float* __restrict__ out,
                           int n, int p, int stride, int total) {
  int idx = blockIdx.x * blockDim.x + threadIdx.x;
  if (idx >= total) return;
  int j = idx % p;
  int r = idx / p;
  int i = r % n;
  int b = r / n;
  int t = i * stride + j;
  if (t > 4095) t = 4095;   // pad with last sample (all pad copies equal y[:,4095])
  out[idx] = y[b * 4096 + t];
}

// ---------- LayerNorm (unbiased std, /(std+eps)); one wave per row, d in {32,64} ----------
__global__ void mtst_ln(const float* __restrict__ x, const float* __restrict__ g,
                        const float* __restrict__ bt, float* __restrict__ o, int d) {
  int row = blockIdx.x;
  int lane = threadIdx.x;
  const float* xr = x + (size_t)row * d;
  float vals[2];
  int cnt = d >> 5;  // 1 or 2
  float s1 = 0.f;
  for (int c = 0; c < cnt; ++c) { vals[c] = xr[lane + 32 * c]; s1 += vals[c]; }
  s1 = wred_sum32(s1);
  float mean = s1 / (float)d;
  float s2 = 0.f;
  for (int c = 0; c < cnt; ++c) { float dv = vals[c] - mean; s2 += dv * dv; }
  s2 = wred_sum32(s2);
  float stdv = sqrtf(s2 / (float)(d - 1));     // ddof=1
  float inv = 1.f / (stdv + 1e-6f);
  float* orow = o + (size_t)row * d;
  for (int c = 0; c < cnt; ++c) {
    int e = lane + 32 * c;
    orow[e] = g[e] * (vals[c] - mean) * inv + bt[e];
  }
}

// ---------- generic fp32 WMMA GEMM: C = [relu?]( A[M,K]*B[K,N] + bias + R ) ----------
// One wave per 16x16 tile. M%16==0, N%16==0, K%4==0 guaranteed by caller.
// A-frag layout (16x4 f32): lane L -> m=L&15, K pair = (L>>4)*2. B mirrored. C/D: 8 VGPRs.
__global__ void mtst_gemm16(const float* __restrict__ A, const float* __restrict__ Bm,
                            const float* __restrict__ bias, const float* __restrict__ R,
                            float* __restrict__ C, int M, int N, int K, int relu) {
  int m0 = blockIdx.x * 16;
  int n0 = blockIdx.y * 16;
  int lane = threadIdx.x;
  int hi = lane >> 4;         // 0 or 1
  int lm = lane & 15;
  int kb = hi * 2;            // K sub-offset 0 or 2
  const float* arow = A + (size_t)(m0 + lm) * K + kb;
  const float* bcol = Bm + (size_t)kb * N + n0 + lm;
  v8f acc = {};
  for (int k = 0; k < K; k += 4) {
    if ((k & 15) == 0) {      // uniform scalar condition: no EXEC divergence
      __builtin_prefetch(&arow[k + 16], 0, 3);
      __builtin_prefetch(&bcol[(size_t)(k + 16) * N], 0, 3);
    }
    v2f a;  a.x = arow[k];              a.y = arow[k + 1];
    v2f bb; bb.x = bcol[(size_t)k * N]; bb.y = bcol[(size_t)(k + 1) * N];
    acc = wmma_f32(a, bb, acc);
  }
  int n = n0 + lm;
  float bv = bias ? bias[n] : 0.f;
#pragma unroll
  for (int i = 0; i < 8; ++i) {
    int m = m0 + i + hi * 8;
    float v = acc[i] + bv;
    if (R) v += R[(size_t)m * N + n];
    if (relu) v = fmaxf(v, 0.f);
    C[(size_t)m * N + n] = v;
  }
}

// ---------- fused relative-attention (flash softmax + skew), one wave per (b,h,16 rows) ----------
// skew(QEr)[s,t] = q[s].Er[S-1+t-s]  (t<=s) ; 0 (t==s+1) ; q[s+1].Er[t-s-2] (t>s+1)
__global__ void mtst_attn(const float* __restrict__ q, const float* __restrict__ k,
                          const float* __restrict__ v, const float* __restrict__ Er,
                          float* __restrict__ out, int S, int d, int dh, float scale) {
  extern __shared__ float qer[];   // 17 * S floats: QEr rows s0..s0+16
  int sb = blockIdx.x, h = blockIdx.y, b = blockIdx.z;
  int s0 = sb * 16;
  int lane = threadIdx.x;
  int hi = lane >> 4;
  int lm = lane & 15;
  const size_t bS = (size_t)b * S;

  // 1) QEr block -> LDS (VALU; dh<=4 so this is cheap)
  int tot = 17 * S;
  for (int idx = lane; idx < tot; idx += 32) {
    int r = idx / S, t = idx - r * S;
    int srow = s0 + r; if (srow > S - 1) srow = S - 1;
    const float* qp = q + (bS + srow) * d + h * dh;
    const float* ep = Er + (size_t)t * dh;
    float a = 0.f;
    for (int dd = 0; dd < dh; ++dd) a += qp[dd] * ep[dd];
    qer[idx] = a;
  }
  __syncthreads();

  // 2) Q fragment (A-layout, K = dh zero-padded to 4)
  int kb = hi * 2;
  int qm = s0 + lm; if (qm > S - 1) qm = S - 1;
  const float* qp = q + (bS + qm) * d + h * dh;
  v2f aq;
  aq.x = (kb     < dh) ? qp[kb]     : 0.f;
  aq.y = (kb + 1 < dh) ? qp[kb + 1] : 0.f;

  // 3) flash loop over key tiles
  float runm[8], lsum[8], acc[8][4];
#pragma unroll
  for (int i = 0; i < 8; ++i) {
    runm[i] = -3.0e38f; lsum[i] = 0.f;
    for (int dd = 0; dd < 4; ++dd) acc[i][dd] = 0.f;
  }
  int ntiles = (S + 15) >> 4;
  for (int tt = 0; tt < ntiles; ++tt) {
    int t0 = tt * 16;
    int tn = t0 + lm;
    v2f bk; bk.x = 0.f; bk.y = 0.f;
    float vv[4] = {0.f, 0.f, 0.f, 0.f};
    if (tn < S) {
      const float* kp = k + (bS + tn) * d + h * dh;
      if (kb     < dh) bk.x = kp[kb];
      if (kb + 1 < dh) bk.y = kp[kb + 1];
      const float* vp = v + (bS + tn) * d + h * dh;
      for (int dd = 0; dd < dh; ++dd) vv[dd] = vp[dd];
    }
    v8f z = {};
    v8f sc = wmma_f32(aq, bk, z);     // EXEC all-ones here (reconverged)
#pragma unroll
    for (int i = 0; i < 8; ++i) {
      int m = i + hi * 8;
      int s = s0 + m;
      int t = tn;
      float sk;
      if (t == s + 1) {
        sk = 0.f;
      } else if (t <= s) {
        int col = S - 1 + t - s;           // can underflow only on discarded rows
        col = col < 0 ? 0 : col;
        sk = qer[m * S + col];
      } else {
        sk = qer[(m + 1) * S + (t - s - 2)];
      }
      float val = (sc[i] + sk) * scale;
      if (t >= S) val = -3.0e38f;          // mask padded keys
      float tmax = nred_max(val);
      float newm = fmaxf(runm[i], tmax);
      float f = __expf(runm[i] - newm);
      float p = __expf(val - newm);
      lsum[i] = lsum[i] * f + p;
      for (int dd = 0; dd < 4; ++dd) acc[i][dd] = acc[i][dd] * f + p * vv[dd];
      runm[i] = newm;
    }
  }

  // 4) reduce partial sums across the 16 N-lanes, write rows
#pragma unroll
  for (int i = 0; i < 8; ++i) {
    lsum[i] = nred_sum(lsum[i]);
    for (int dd = 0; dd < 4; ++dd) acc[i][dd] = nred_sum(acc[i][dd]);
  }
  if (lm == 0) {
    for (int i = 0; i < 8; ++i) {
      int s = s0 + i + hi * 8;
      if (s < S) {
        float invl = 1.f / lsum[i];
        float* op = out + (bS + s) * d + h * dh;
        for (int dd = 0; dd < dh; ++dd) op[dd] = acc[i][dd] * invl;
      }
    }
  }
}

// ---------- final FC over concatenated branch activations; M=16, WMMA K-loop ----------
// K split into two clean segments (A0 rows, then A1 rows); W_ff streamed with prefetch.
__global__ void mtst_finalfc(const float* __restrict__ A0, const float* __restrict__ A1,
                             const float* __restrict__ W, const float* __restrict__ bias,
                             float* __restrict__ out, int K0, int K, int N) {
  int n0 = blockIdx.x * 16;
  int lane = threadIdx.x;
  int hi = lane >> 4;
  int lm = lane & 15;
  int kb = hi * 2;
  int K1 = K - K0;
  const float* a0 = A0 + (size_t)lm * K0 + kb;
  const float* a1 = A1 + (size_t)lm * K1 + kb;
  const float* w0 = W + (size_t)kb * N + n0 + lm;
  const float* w1 = w0 + (size_t)K0 * N;
  v8f acc = {};
  for (int kk = 0; kk < K0; kk += 4) {
    __builtin_prefetch(&w0[(size_t)(kk + 128) * N], 0, 0);  // stream W_ff ahead
    v2f a;  a.x = a0[kk];              a.y = a0[kk + 1];
    v2f bb; bb.x = w0[(size_t)kk * N]; bb.y = w0[(size_t)(kk + 1) * N];
    acc = wmma_f32(a, bb, acc);
  }
  for (int kk = 0; kk < K1; kk += 4) {
    __builtin_prefetch(&w1[(size_t)(kk + 128) * N], 0, 0);
    v2f a;  a.x = a1[kk];              a.y = a1[kk + 1];
    v2f bb; bb.x = w1[(size_t)kk * N]; bb.y = w1[(size_t)(kk + 1) * N];
    acc = wmma_f32(a, bb, acc);
  }
  int n = n0 + lm;
  float bv = bias[n];
#pragma unroll
  for (int i = 0; i < 8; ++i) {
    int m = i + hi * 8;
    out[(size_t)m * N + n] = acc[i] + bv;
  }
}

// ---------- host orchestration ----------
enum { P_Wq, P_bq, P_Wk, P_bk, P_Wv, P_bv, P_Er, P_Wo, P_bo,
       P_w1, P_b1, P_w2, P_b2, P_l1a, P_l1b, P_l2a, P_l2b };

extern "C" void kernel_launch(void* const* d_in, const int* in_sizes, int n_in,
                              void* d_out, int out_size, void* d_ws, size_t ws_size,
                              hipStream_t stream) {
  (void)out_size; (void)ws_size;
  // Input-order detection: insertion order puts y (65536) first; jax alphabetical
  // tree-flatten puts W_ff (4,169,728) first.
  bool alpha = (n_in > 0 && in_sizes[0] != 65536);
  static const int offIns[17]   = {0,1,2,3,4,5,6,7,8,9,10,11,12,13,14,15,16};
  static const int offAlpha[17] = {3,9,1,7,4,10,0,2,8,15,5,16,6,11,12,13,14};
  const int* off = alpha ? offAlpha : offIns;
  const int yIdx   = alpha ? 108 : 0;
  const int wffIdx = alpha ? 0   : 107;
  const int bffIdx = alpha ? 1   : 108;
  const int brBase = alpha ? 2   : 1;

  const float* y = (const float*)d_in[yIdx];

  // workspace layout (floats)
  const size_t RMAX = (size_t)16 * 510 * 32;        // 261120 (>= 16*254*64)
  float* ws   = (float*)d_ws;
  float* xn   = ws;                                  // LN output
  float* qb   = xn + RMAX;
  float* kbuf = qb + RMAX;
  float* vb   = kbuf + RMAX;
  float* ao   = vb + RMAX;
  float* h1   = ao + RMAX;                           // FFN hidden: 16*510*512 max
  float* x0   = h1 + (size_t)16 * 510 * 512;
  float* x1   = x0 + (size_t)16 * 510 * 32;

  const int PS[2] = {32, 64}, ST[2] = {8, 16}, NP[2] = {510, 254};
  float* xb[2] = {x0, x1};

  for (int br = 0; br < 2; ++br) {
    const int p = PS[br], stp = ST[br], n = NP[br];
    const int d = p, S = n, dh = d / 16;             // H = 16 heads
    const int BS = 16 * S;                           // rows (batch*seq)
    float* x = xb[br];
    const int base = brBase + br * 53;

    int total = 16 * n * p;
    mtst_patch<<<(total + 255) / 256, 256, 0, stream>>>(y, x, n, p, stp, total);

    for (int L = 0; L < 3; ++L) {
      const int lb = base + L * 17;
      #define LP(e) ((const float*)d_in[lb + off[e]])
      const float *Wq = LP(P_Wq), *bq = LP(P_bq), *Wk = LP(P_Wk), *bk = LP(P_bk);
      const float *Wv = LP(P_Wv), *bv = LP(P_bv), *Er = LP(P_Er);
      const float *Wo = LP(P_Wo), *bo = LP(P_bo);
      const float *w1 = LP(P_w1), *b1 = LP(P_b1), *w2 = LP(P_w2), *b2 = LP(P_b2);
      const float *l1a = LP(P_l1a), *l1b = LP(P_l1b), *l2a = LP(P_l2a), *l2b = LP(P_l2b);
      #undef LP

      mtst_ln<<<BS, 32, 0, stream>>>(x, l1a, l1b, xn, d);

      dim3 gq(BS / 16, d / 16);
      mtst_gemm16<<<gq, 32, 0, stream>>>(xn, Wq, bq, nullptr, qb,   BS, d, d, 0);
      mtst_gemm16<<<gq, 32, 0, stream>>>(xn, Wk, bk, nullptr, kbuf, BS, d, d, 0);
      mtst_gemm16<<<gq, 32, 0, stream>>>(xn, Wv, bv, nullptr, vb,   BS, d, d, 0);

      dim3 ga((S + 15) / 16, 16, 16);
      size_t shmem = (size_t)17 * S * sizeof(float);
      float scale = 1.0f / sqrtf((float)dh);
      mtst_attn<<<ga, 32, shmem, stream>>>(qb, kbuf, vb, Er, ao, S, d, dh, scale);

      // x = x + attn_out @ Wo + bo   (in-place residual, element-private)
      mtst_gemm16<<<gq, 32, 0, stream>>>(ao, Wo, bo, x, x, BS, d, d, 0);

      mtst_ln<<<BS, 32, 0, stream>>>(x, l2a, l2b, xn, d);

      dim3 g1(BS / 16, 512 / 16);
      mtst_gemm16<<<g1, 32, 0, stream>>>(xn, w1, b1, nullptr, h1, BS, 512, d, 1);
      // x = x + relu(h1) @ w2 + b2
      mtst_gemm16<<<gq, 32, 0, stream>>>(h1, w2, b2, x, x, BS, d, 512, 0);
    }

    const float* na = (const float*)d_in[base + 51];
    const float* nb = (const float*)d_in[base + 52];
    mtst_ln<<<BS, 32, 0, stream>>>(x, na, nb, x, d);   // in-place final norm
  }

  const float* Wff = (const float*)d_in[wffIdx];
  const float* bff = (const float*)d_in[bffIdx];
  // K0 = 510*32 = 16320, K = 16320 + 254*64 = 32576, N = 128
  mtst_finalfc<<<dim3(128 / 16), 32, 0, stream>>>(x0, x1, Wff, bff, (float*)d_out,
                                                  16320, 32576, 128);
}